// RoiPooling_18794776887350
// MI455X (gfx1250) — compile-verified
//
#include <hip/hip_runtime.h>

typedef float v2f __attribute__((ext_vector_type(2)));
typedef float v8f __attribute__((ext_vector_type(8)));
typedef int   v4i __attribute__((__vector_size__(16)));   // matches builtin's param type

#define H_    200
#define W_    304
#define C_    256
#define HP_   202          // symmetric pad 1
#define WP_   306
#define POOL_ 7
#define S_    14
#define OUTPER_ (C_*POOL_*POOL_)   // 12544 floats per ROI

// ---- CDNA5 async global->LDS copy (ASYNCcnt path) ----
__device__ __forceinline__ void async_b128(const float* gsrc, float* ldst) {
#if __has_builtin(__builtin_amdgcn_global_load_async_to_lds_b128)
  float* g = const_cast<float*>(gsrc);
  __builtin_amdgcn_global_load_async_to_lds_b128(
      (__attribute__((address_space(1))) v4i*)g,
      (__attribute__((address_space(3))) v4i*)ldst, 0, 0);
#else
  unsigned lds_off = (unsigned)(unsigned long long)
      (__attribute__((address_space(3))) char*)ldst;
  asm volatile("global_load_async_to_lds_b128 %0, %1, off"
               :: "v"(lds_off), "v"(gsrc) : "memory");
#endif
}

__device__ __forceinline__ void wait_async0() {
#if __has_builtin(__builtin_amdgcn_s_wait_asynccnt)
  __builtin_amdgcn_s_wait_asynccnt(0);
#else
  asm volatile("s_wait_asynccnt 0x0" ::: "memory");
#endif
}

// Base A weight (ROI-constant): pooled cell m takes taps j=2m,2m+1.
// kk in [0,28): kk<14 -> (1-lx) tap at colIdx[kk], kk>=14 -> lx tap at colIdx[kk].
// Folds the 2x2-pool 0.25 and column validity vx.
__device__ __forceinline__ float bw(int m, int kk, const float* lx, const float* vx) {
  const int   j = (kk < 14) ? kk : kk - 14;
  const float h = (kk < 14) ? (1.0f - lx[j]) : lx[j];
  const float w = 0.25f * vx[j] * h;
  return ((j >> 1) == m) ? w : 0.0f;
}

__global__ __launch_bounds__(256)
void roi_align_wmma(const float* __restrict__ fmap,
                    const float* __restrict__ rois,
                    const int*   __restrict__ stridePtr,
                    float* __restrict__ out) {
  __shared__ float s_raw[56 * C_];       // 57344 B : K-major, K = r*28 + jj
  __shared__ float s_pooled[OUTPER_];    // 50176 B : NCHW-ordered staging
  __shared__ int   s_colIdx[28];
  __shared__ int   s_rowA[S_], s_rowB[S_];
  __shared__ float s_lx[S_], s_vx[S_], s_ly[S_], s_vy[S_];

  const int n = blockIdx.x;
  const int t = threadIdx.x;

  // ---- per-ROI sampling metadata (matches reference idx()) ----
  if (t < 28) {
    const bool  isY = t < 14;
    const int   idx = isY ? t : t - 14;
    const float inv = 1.0f / (float)(*stridePtr);
    const float lo  = rois[n*4 + (isY ? 1 : 0)] * inv + 1.0f;   // padded coords
    const float hi  = rois[n*4 + (isY ? 3 : 2)] * inv + 1.0f;
    const float sc  = (hi - lo) * (1.0f / 14.0f);
    const float coord = lo + sc * ((float)idx + 0.5f) - 0.5f;
    const int size   = isY ? HP_ : WP_;
    const int rawMax = isY ? (H_-1) : (W_-1);
    const float f0   = floorf(coord);
    const float frac = coord - f0;
    int i0 = (int)f0; i0 = min(max(i0, 0), size-1);
    const int i1 = min(i0 + 1, size-1);
    const float valid = (coord >= 0.0f && coord <= (float)(size-1)) ? 1.0f : 0.0f;
    const int m0 = min(max(i0-1, 0), rawMax);   // symmetric pad(1) == clamp
    const int m1 = min(max(i1-1, 0), rawMax);
    if (isY) { s_rowA[idx]=m0; s_rowB[idx]=m1; s_ly[idx]=frac; s_vy[idx]=valid; }
    else     { s_colIdx[idx]=m0; s_colIdx[14+idx]=m1; s_lx[idx]=frac; s_vx[idx]=valid; }
  }
  __syncthreads();

  const int lane = t & 31;
  const int wave = t >> 5;
  const int hi2  = (lane >= 16) ? 2 : 0;   // ISA A layout: hi half holds K+2,K+3
  const int M    = lane & 15;
  const int cc0  = wave * 2;               // wave owns channel chunks cc0, cc0+1

  // ---- precompute per-lane A base once per ROI (ks and ks+7 share it) ----
  v2f baseA[7];
  #pragma unroll
  for (int ks = 0; ks < 7; ++ks) {
    const int k0 = ks*4 + hi2;
    baseA[ks].x = bw(M, k0,   s_lx, s_vx);
    baseA[ks].y = bw(M, k0+1, s_lx, s_vx);
  }

  for (int pr = 0; pr < POOL_; ++pr) {
    v8f acc0 = {0,0,0,0,0,0,0,0};
    v8f acc1 = {0,0,0,0,0,0,0,0};
    for (int ii = 0; ii < 2; ++ii) {
      const int i  = pr*2 + ii;
      const int r0 = s_rowA[i];
      const int r1 = s_rowB[i];
      // ---- pure async copy: 2 rows x 28 cols x 256 ch -> s_raw (b128/lane) ----
      #pragma unroll
      for (int it2 = 0; it2 < 14; ++it2) {
        const int r   = (it2 >= 7) ? 1 : 0;
        const int w2  = t + (it2 - 7*r)*256;   // 0..1791 == 28 cols * 64 float4
        const int jj  = w2 >> 6;
        const int c4  = w2 & 63;
        const int row = r ? r1 : r0;
        const int col = s_colIdx[jj];
        const float* gsrc = fmap + ((size_t)row*W_ + col)*C_ + (size_t)c4*4;
        float*       ldst = s_raw + ((r*28 + jj)*64 + c4)*4;
        async_b128(gsrc, ldst);
      }
      wait_async0();
      __syncthreads();
      // ---- K=56 WMMA: vertical lerp folded into per-kstep A scale ----
      {
        const float vyi = s_vy[i], lyi = s_ly[i];
        const float sy0 = vyi * (1.0f - lyi);   // row0 taps
        const float sy1 = vyi * lyi;            // row1 taps
        #pragma unroll
        for (int ks = 0; ks < 14; ++ks) {
          const float sf = (ks < 7) ? sy0 : sy1;
          const v2f  ab  = baseA[(ks < 7) ? ks : ks - 7];
          v2f a; a.x = ab.x * sf; a.y = ab.y * sf;
          const int k0 = ks*4 + hi2;
          v2f b;
          b.x = s_raw[(k0  )*C_ + cc0*16 + M];
          b.y = s_raw[(k0+1)*C_ + cc0*16 + M];
          acc0 = __builtin_amdgcn_wmma_f32_16x16x4_f32(false, a, false, b,
                                                       (short)0, acc0, false, false);
          v2f b2;
          b2.x = s_raw[(k0  )*C_ + cc0*16 + 16 + M];
          b2.y = s_raw[(k0+1)*C_ + cc0*16 + 16 + M];
          acc1 = __builtin_amdgcn_wmma_f32_16x16x4_f32(false, a, false, b2,
                                                       (short)0, acc1, false, false);
        }
      }
      __syncthreads();
    }
    // ---- D layout: VGPR r = pooled col r (lanes 0-15); rows 7..15 zero ----
    if (lane < 16) {
      const int c0 = (cc0  )*16 + lane;
      const int c1 = (cc0+1)*16 + lane;
      #pragma unroll
      for (int r = 0; r < 7; ++r) {
        s_pooled[c0*49 + pr*7 + r] = acc0[r];
        s_pooled[c1*49 + pr*7 + r] = acc1[r];
      }
    }
  }
  __syncthreads();
  // ---- fully coalesced b128 store of the NCHW-flattened ROI ----
  float4*       op = (float4*)(out + (size_t)n * OUTPER_);
  const float4* pp = (const float4*)s_pooled;
  for (int idx = t; idx < OUTPER_/4; idx += 256) op[idx] = pp[idx];
}

extern "C" void kernel_launch(void* const* d_in, const int* in_sizes, int n_in,
                              void* d_out, int out_size, void* d_ws, size_t ws_size,
                              hipStream_t stream) {
  const float* fmap      = (const float*)d_in[0];
  const float* rois      = (const float*)d_in[1];
  const int*   stridePtr = (const int*)d_in[2];
  float*       out       = (float*)d_out;
  const int N = in_sizes[1] / 4;   // rois is [N,4]
  roi_align_wmma<<<N, 256, 0, stream>>>(fmap, rois, stridePtr, out);
}